// RegressionFNP_78374563217654
// MI455X (gfx1250) — compile-verified
//
#include <hip/hip_runtime.h>
#include <hip/hip_bf16.h>

// ---------------------------------------------------------------------------
// FNP forward for MI455X (gfx1250, wave32, WMMA).
// - fuse G/A adjacency construction with GA @ [Vmean|Vlogscale] (never
//   materialize the 2x64MB G/A matrices; only mandatory HBM traffic is the
//   one-pass 128MB U_dag/U_A stream, ~5.5us at 23.3 TB/s)
// - G half runs in sorted coordinates: U_dag streams coalesced, triu mask
//   becomes p<q so the lower triangle is skipped entirely
// - all dense GEMM / dot work on V_WMMA_F32_16X16X4_F32 (native fp32 matrix
//   path; data is fp32 and the problem is not matmul-bound)
// - fused kernel runs at 1 workgroup/CU (__launch_bounds__(128,1)) so the
//   query tile + 80 accumulators stay in VGPRs (round 2 spilled to scratch)
// ---------------------------------------------------------------------------

typedef __attribute__((ext_vector_type(2))) float v2f;
typedef __attribute__((ext_vector_type(8))) float v8f;

#define NRR 4096
#define NALL 8192
#define DXX 128
#define DHH 512
#define DUU 64

__device__ __forceinline__ v8f wmma_f32x4(v2f a, v2f b, v8f c) {
  // D(16x16,f32) += A(16x4,f32) * B(4x16,f32)
  return __builtin_amdgcn_wmma_f32_16x16x4_f32(
      /*neg_a=*/false, a, /*neg_b=*/false, b,
      /*c_mod=*/(short)0, c, /*reuse_a=*/false, /*reuse_b=*/false);
}

// ---------------------------------------------------------------------------
// fp32 WMMA GEMM: C[M,N] = act(A @ B + bias)
// MODE 0: A = A0, row-major, ld = K
// MODE 1: A rows are concat(XR, XM), ld = 128 (rows >= 4096 from A1)
// MODE 2: A cols are concat(Z(64), U(64)), ld = 64 each (final_rep)
// One wave per 16x16 tile; block = 4 waves = 16 rows x 64 cols.
// ---------------------------------------------------------------------------
template <int MODE, int RELU>
__global__ __launch_bounds__(128) void k_gemm_wmma(
    const float* __restrict__ A0, const float* __restrict__ A1,
    const float* __restrict__ B, const float* __restrict__ bias,
    float* __restrict__ C, int M, int N, int K) {
  int lane = threadIdx.x & 31;
  int wave = threadIdx.x >> 5;
  int n0 = blockIdx.x * 64 + wave * 16;
  int m0 = blockIdx.y * 16;
  if (n0 >= N || m0 >= M) return;     // wave-uniform: EXEC stays full

  int half = (lane < 16) ? 0 : 2;     // K sub-offset for this half-wave
  int rq   = lane & 15;               // A row within tile / B,C column
  int arow = m0 + rq;

  // Loop-invariant source pointers (hoisted row select for MODE 1).
  const float* ap;
  if (MODE == 1) {
    ap = (arow < NRR) ? (A0 + (size_t)arow * 128)
                      : (A1 + (size_t)(arow - NRR) * 128);
  } else if (MODE == 2) {
    ap = A0 + (size_t)arow * 64;
  } else {
    ap = A0 + (size_t)arow * K;
  }
  ap += half;
  const float* bp = B + (size_t)half * N + n0 + rq;
  const size_t bstep = (size_t)4 * N;

  v8f c = {};
  if (MODE == 2) {
    const float* ap2 = A1 + (size_t)arow * 64 + half;
#pragma unroll 4
    for (int kb = 0; kb < 64; kb += 4) {
      v2f a = *(const v2f*)ap; ap += 4;
      v2f b; b.x = bp[0]; b.y = bp[N]; bp += bstep;
      c = wmma_f32x4(a, b, c);
    }
#pragma unroll 4
    for (int kb = 0; kb < 64; kb += 4) {
      v2f a = *(const v2f*)ap2; ap2 += 4;
      v2f b; b.x = bp[0]; b.y = bp[N]; bp += bstep;
      c = wmma_f32x4(a, b, c);
    }
  } else {
#pragma unroll 4
    for (int kb = 0; kb < K; kb += 4) {
      v2f a = *(const v2f*)ap; ap += 4;
      v2f b; b.x = bp[0]; b.y = bp[N]; bp += bstep;
      c = wmma_f32x4(a, b, c);
    }
  }

  float bv = bias ? bias[n0 + rq] : 0.0f;
  int rbase = m0 + ((lane < 16) ? 0 : 8);
#pragma unroll
  for (int v = 0; v < 8; ++v) {
    float x = c[v] + bv;
    if (RELU) x = fmaxf(x, 0.0f);
    C[(size_t)(rbase + v) * N + n0 + rq] = x;
  }
}

// ---------------------------------------------------------------------------
// Elementwise: u = pu_mean + exp(pu_ls)*eps_u ; z likewise.
// Rows < NR also produce Vm = cy_mean + qz_mean, Vls = cy_ls + qz_ls.
// ---------------------------------------------------------------------------
__global__ void k_sample(const float* __restrict__ pu,
                         const float* __restrict__ qz,
                         const float* __restrict__ eps_u,
                         const float* __restrict__ eps_z,
                         const float* __restrict__ yR,
                         const float* __restrict__ W_cy,
                         const float* __restrict__ b_cy,
                         float* __restrict__ u, float* __restrict__ z,
                         float* __restrict__ vm, float* __restrict__ vls) {
  int idx = blockIdx.x * blockDim.x + threadIdx.x;
  if (idx >= NALL * DUU) return;
  int i = idx >> 6, d = idx & 63;
  float um = pu[(size_t)i * 128 + d], uls = pu[(size_t)i * 128 + 64 + d];
  u[idx] = um + expf(uls) * eps_u[idx];
  float qm = qz[(size_t)i * 128 + d], qls = qz[(size_t)i * 128 + 64 + d];
  z[idx] = qm + expf(qls) * eps_z[idx];
  if (i < NRR) {
    float y = yR[i];
    vm[idx]  = y * W_cy[d]      + b_cy[d]      + qm;
    vls[idx] = y * W_cy[64 + d] + b_cy[64 + d] + qls;
  }
}

// Row squared norms for all 8192 u rows; Gaussian log-CDF sum for rows < NR.
__global__ void k_rownorm(const float* __restrict__ u,
                          float* __restrict__ ra, float* __restrict__ lc) {
  int i = blockIdx.x;
  int lane = threadIdx.x;      // 32 threads
  float s = 0.0f, l = 0.0f;
  for (int d = lane; d < DUU; d += 32) {
    float x = u[(size_t)i * DUU + d];
    s += x * x;
    if (i < NRR) l += logf(0.5f + 0.5f * erff(x * 0.70710678118654752f));
  }
#pragma unroll
  for (int off = 16; off > 0; off >>= 1) {
    s += __shfl_down(s, off, 32);
    l += __shfl_down(l, off, 32);
  }
  if (lane == 0) { ra[i] = s; if (i < NRR) lc[i] = l; }
}

// O(N^2) stable rank -> argsort. sidx[rank[a]] = a  (rank = inv permutation).
__global__ void k_rank(const float* __restrict__ lc, int* __restrict__ sidx) {
  __shared__ int red[256];
  int a = blockIdx.x;
  float v = lc[a];
  int cnt = 0;
  for (int b = threadIdx.x; b < NRR; b += 256) {
    float w = lc[b];
    cnt += (w < v || (w == v && b < a)) ? 1 : 0;
  }
  red[threadIdx.x] = cnt;
  __syncthreads();
  for (int s = 128; s > 0; s >>= 1) {
    if (threadIdx.x < s) red[threadIdx.x] += red[threadIdx.x + s];
    __syncthreads();
  }
  if (threadIdx.x == 0) sidx[red[0]] = a;
}

// Gather sorted-order copies: Y = uR[sort], Vm_s/Vls_s, row norms.
__global__ void k_gather(const int* __restrict__ sidx,
                         const float* __restrict__ u,
                         const float* __restrict__ vm,
                         const float* __restrict__ vls,
                         const float* __restrict__ ra,
                         float* __restrict__ ys, float* __restrict__ vms,
                         float* __restrict__ vlss, float* __restrict__ rbs) {
  int p = blockIdx.x, d = threadIdx.x;   // 64 threads
  int s = sidx[p];
  ys[(size_t)p * 64 + d]   = u[(size_t)s * 64 + d];
  vms[(size_t)p * 64 + d]  = vm[(size_t)s * 64 + d];
  vlss[(size_t)p * 64 + d] = vls[(size_t)s * 64 + d];
  if (d == 0) rbs[p] = ra[s];
}

// ---------------------------------------------------------------------------
// Fused adjacency + GEMM: for a 16-row tile of GA, stream 16-col tiles:
//   dots via WMMA (K=64) -> dist -> logitexp -> +log-odds noise -> sigmoid
//   -> accumulate rowsum, S_mean = GA@Vm, S_ls = GA@Vls in registers.
// TRI=1 (G, sorted coords): only tj >= ti, mask p<q on the diagonal, output
// rows scattered through rowmap=sort_idx. TRI=0 (A): full row, output row =
// rowoff + p.  4 waves/block split the column tiles; partials combined in
// LDS, then a single plain store per output row (no atomics).
// ---------------------------------------------------------------------------
#define LDSTR 17
template <int TRI>
__global__ __launch_bounds__(128, 1) void k_ga_acc(
    const float* __restrict__ Q, const float* __restrict__ Kmat,
    const float* __restrict__ qn, const float* __restrict__ kn,
    const float* __restrict__ Unoise, const float* __restrict__ Vm,
    const float* __restrict__ Vl, const int* __restrict__ rowmap, int rowoff,
    const float* __restrict__ glogscale, float* __restrict__ Smean,
    float* __restrict__ Sls, float* __restrict__ Rsum) {
  __shared__ float lds_g[4][16 * LDSTR];
  __shared__ float lds_acc[4][16 * 129];

  int lane = threadIdx.x & 31;
  int wave = threadIdx.x >> 5;
  int ti = blockIdx.x;
  int p0 = ti * 16;
  int half  = (lane < 16) ? 0 : 2;
  int half8 = (lane < 16) ? 0 : 8;
  int rq = lane & 15;

  float inv_scale = expf(-glogscale[0]);   // 1/scale
  const float inv_temp = 1.0f / 0.3f;

  // Query tile (16x64) resident in registers for the whole column sweep.
  v2f areg[16];
#pragma unroll
  for (int kb = 0; kb < 16; ++kb)
    areg[kb] = *(const v2f*)(Q + (size_t)(p0 + rq) * 64 + kb * 4 + half);
  float qnr[8];
#pragma unroll
  for (int v = 0; v < 8; ++v) qnr[v] = qn[p0 + v + half8];

  float accm[32], accl[32], accr[16];
#pragma unroll
  for (int i = 0; i < 32; ++i) { accm[i] = 0.0f; accl[i] = 0.0f; }
#pragma unroll
  for (int i = 0; i < 16; ++i) accr[i] = 0.0f;

  // One base pointer into the noise stream; per-row deltas v*4096 floats are
  // compile-time constants that fold into the 24-bit instruction offset.
  const float* up = Unoise + (size_t)(p0 + half8) * 4096 + rq;

  for (int tj = (TRI ? ti : 0) + wave; tj < 256; tj += 4) {
    int q0 = tj * 16;
    // Prefetch the next column tile of the 128MB noise stream (gfx1250
    // global_prefetch path) so HBM latency hides under the transcendentals.
    if (tj + 4 < 256) {
#pragma unroll
      for (int v = 0; v < 8; ++v)
        __builtin_prefetch(up + (size_t)v * 4096 + q0 + 64, 0, 1);
    }
    // --- dot tile: preload B tile, then 16 x V_WMMA_F32_16X16X4_F32 ---
    v2f breg[16];
    {
      const float* bp = Kmat + (size_t)(q0 + rq) * 64 + half;
#pragma unroll
      for (int kb = 0; kb < 16; ++kb) breg[kb] = *(const v2f*)(bp + kb * 4);
    }
    v8f d = {};
#pragma unroll
    for (int kb = 0; kb < 16; ++kb) d = wmma_f32x4(areg[kb], breg[kb], d);

    float knv = kn[q0 + rq];
    // --- elementwise: dist -> logitexp -> noise -> sigmoid ---
#pragma unroll
    for (int v = 0; v < 8; ++v) {
      float un = up[(size_t)v * 4096 + q0];       // coalesced stream
      float dist = fmaxf(qnr[v] + knv - 2.0f * d[v], 0.0f);
      float logp = -0.5f * dist * inv_scale;
      float logit;
      if (logp <= -0.6931471805599453f)
        logit = logp - log1pf(-expf(logp));
      else
        logit = -logf(fmaxf(expm1f(-logp), 1e-20f));
      float L = logf(un) - log1pf(-un);
      float g = 1.0f / (1.0f + expf(-(logit + L) * inv_temp));
      if (TRI && tj == ti)                        // diagonal tile: strict triu
        g = ((v + half8) < rq) ? g : 0.0f;
      lds_g[wave][rq * LDSTR + v + half8] = g;    // transposed [q][p], padded
    }
    asm volatile("s_wait_dscnt 0" ::: "memory");  // per-wave LDS visibility
    // --- accumulate S and rowsum: lane owns output cols {2*lane, 2*lane+1} ---
#pragma unroll 2
    for (int q = 0; q < 16; ++q) {
      v2f vm2 = *(const v2f*)(Vm + (size_t)(q0 + q) * 64 + 2 * lane);
      v2f vl2 = *(const v2f*)(Vl + (size_t)(q0 + q) * 64 + 2 * lane);
      const float* grow = &lds_g[wave][q * LDSTR];
#pragma unroll
      for (int p = 0; p < 16; ++p) {
        float g = grow[p];                        // LDS broadcast
        accm[2 * p]     += g * vm2.x;
        accm[2 * p + 1] += g * vm2.y;
        accl[2 * p]     += g * vl2.x;
        accl[2 * p + 1] += g * vl2.y;
        accr[p]         += g;                     // identical in all lanes
      }
    }
  }

  // Combine 4 wave-partials in LDS; one plain store per output row element.
#pragma unroll
  for (int p = 0; p < 16; ++p) {
    lds_acc[wave][p * 129 + 2 * lane]          = accm[2 * p];
    lds_acc[wave][p * 129 + 2 * lane + 1]      = accm[2 * p + 1];
    lds_acc[wave][p * 129 + 64 + 2 * lane]     = accl[2 * p];
    lds_acc[wave][p * 129 + 64 + 2 * lane + 1] = accl[2 * p + 1];
    if (lane == 0) lds_acc[wave][p * 129 + 128] = accr[p];
  }
  __syncthreads();
  for (int e = threadIdx.x; e < 16 * 129; e += 128) {
    int p = e / 129, c = e - p * 129;
    float s = lds_acc[0][e] + lds_acc[1][e] + lds_acc[2][e] + lds_acc[3][e];
    int orow = rowmap ? rowmap[p0 + p] : (rowoff + p0 + p);
    if (c < 64)       Smean[(size_t)orow * 64 + c] = s;
    else if (c < 128) Sls[(size_t)orow * 64 + (c - 64)] = s;
    else              Rsum[orow] = s;
  }
}

// pqz row sums: normal_logprob(z, pz) - normal_logprob(z, qz). (log 2pi cancels)
__global__ void k_pqz(const float* __restrict__ z, const float* __restrict__ qz,
                      const float* __restrict__ Sm, const float* __restrict__ Sl,
                      const float* __restrict__ rs, float* __restrict__ accum) {
  int i = blockIdx.x, lane = threadIdx.x;   // 32 threads
  float inv = 1.0f / (rs[i] + 1e-8f);
  float s = 0.0f;
  for (int d = lane; d < 64; d += 32) {
    float zz = z[(size_t)i * 64 + d];
    float pm = Sm[(size_t)i * 64 + d] * inv;
    float pl = Sl[(size_t)i * 64 + d] * inv;
    float qm = qz[(size_t)i * 128 + d];
    float ql = qz[(size_t)i * 128 + 64 + d];
    float tp = (zz - pm) * expf(-pl);
    float tq = (zz - qm) * expf(-ql);
    s += (-0.5f * tp * tp - pl) - (-0.5f * tq * tq - ql);
  }
#pragma unroll
  for (int off = 16; off > 0; off >>= 1) s += __shfl_down(s, off, 32);
  if (lane == 0) atomicAdd(&accum[(i < NRR) ? 0 : 1], s);
}

// Output head: out = h @ W_o2 + b_o2 (N=2), then Gaussian log-likelihood.
__global__ void k_outhead(const float* __restrict__ h,
                          const float* __restrict__ W,
                          const float* __restrict__ b,
                          const float* __restrict__ yR,
                          const float* __restrict__ yM,
                          float* __restrict__ accum) {
  int i = blockIdx.x, lane = threadIdx.x;   // 32 threads
  float a0 = 0.0f, a1 = 0.0f;
  for (int k = lane; k < DHH; k += 32) {
    float hv = h[(size_t)i * DHH + k];
    a0 += hv * W[k * 2 + 0];
    a1 += hv * W[k * 2 + 1];
  }
#pragma unroll
  for (int off = 16; off > 0; off >>= 1) {
    a0 += __shfl_down(a0, off, 32);
    a1 += __shfl_down(a1, off, 32);
  }
  if (lane == 0) {
    float m = a0 + b[0];
    float lraw = a1 + b[1];
    float sp = (lraw > 20.0f) ? lraw : log1pf(expf(lraw));
    float lstd = logf(0.1f + 0.9f * sp);
    float y = (i < NRR) ? yR[i] : yM[i - NRR];
    float t = (y - m) * expf(-lstd);
    float lp = -0.5f * t * t - lstd - 0.91893853320467274f;
    atomicAdd(&accum[(i < NRR) ? 2 : 3], lp);
  }
}

__global__ void k_init(float* __restrict__ accum) {
  if (threadIdx.x < 4) accum[threadIdx.x] = 0.0f;
}

__global__ void k_final(const float* __restrict__ accum, float* __restrict__ out) {
  out[0] = -((accum[2] + accum[0]) / 4096.0f + (accum[3] + accum[1]) / 4096.0f);
}

// ---------------------------------------------------------------------------
extern "C" void kernel_launch(void* const* d_in, const int* in_sizes, int n_in,
                              void* d_out, int out_size, void* d_ws, size_t ws_size,
                              hipStream_t stream) {
  (void)in_sizes; (void)n_in; (void)out_size; (void)ws_size;
  const float* XR   = (const float*)d_in[0];
  const float* yR   = (const float*)d_in[1];
  const float* XM   = (const float*)d_in[2];
  const float* yM   = (const float*)d_in[3];
  const float* W_ct = (const float*)d_in[4];
  const float* b_ct = (const float*)d_in[5];
  const float* W_pu = (const float*)d_in[6];
  const float* b_pu = (const float*)d_in[7];
  const float* W_qz = (const float*)d_in[8];
  const float* b_qz = (const float*)d_in[9];
  const float* W_cy = (const float*)d_in[10];
  const float* b_cy = (const float*)d_in[11];
  const float* W_o1 = (const float*)d_in[12];
  const float* b_o1 = (const float*)d_in[13];
  const float* W_o2 = (const float*)d_in[14];
  const float* b_o2 = (const float*)d_in[15];
  const float* g_ls = (const float*)d_in[16];
  const float* eps_u = (const float*)d_in[17];
  const float* eps_z = (const float*)d_in[18];
  const float* U_dag = (const float*)d_in[19];
  const float* U_A   = (const float*)d_in[20];

  float* ws = (float*)d_ws;
  float* H    = ws;                         // 8192*512  (reused for h2 later)
  float* PU   = ws + 4194304;               // 8192*128
  float* QZ   = ws + 5242880;               // 8192*128
  float* UU   = ws + 6291456;               // 8192*64   u samples
  float* ZZ   = ws + 6815744;               // 8192*64   z samples
  float* VM   = ws + 7340032;               // 4096*64
  float* VLS  = ws + 7602176;               // 4096*64
  float* RA   = ws + 7864320;               // 8192      row |u|^2
  float* LC   = ws + 7872512;               // 4096      logcdf
  int*   SI   = (int*)(ws + 7876608);       // 4096      sort_idx
  float* YS   = ws + 7880704;               // 4096*64   sorted u rows
  float* VMS  = ws + 8142848;               // 4096*64
  float* VLSS = ws + 8404992;               // 4096*64
  float* RBS  = ws + 8667136;               // 4096
  float* SM   = ws + 8671232;               // 8192*64
  float* SL   = ws + 9195520;               // 8192*64
  float* RS   = ws + 9719808;               // 8192
  float* ACC  = ws + 9728000;               // 4 scalar accumulators

  k_init<<<1, 32, 0, stream>>>(ACC);

  // H = relu(concat(XR,XM) @ W_ct + b_ct)   [8192x128x512]
  k_gemm_wmma<1, 1><<<dim3(8, 512), 128, 0, stream>>>(XR, XM, W_ct, b_ct, H,
                                                      NALL, DHH, DXX);
  // pu / qz heads   [8192x512x128]
  k_gemm_wmma<0, 0><<<dim3(2, 512), 128, 0, stream>>>(H, nullptr, W_pu, b_pu,
                                                      PU, NALL, 128, DHH);
  k_gemm_wmma<0, 0><<<dim3(2, 512), 128, 0, stream>>>(H, nullptr, W_qz, b_qz,
                                                      QZ, NALL, 128, DHH);
  // samples + V vectors
  k_sample<<<(NALL * DUU) / 256, 256, 0, stream>>>(PU, QZ, eps_u, eps_z, yR,
                                                   W_cy, b_cy, UU, ZZ, VM, VLS);
  // norms + logcdf, rank->argsort, sorted gathers
  k_rownorm<<<NALL, 32, 0, stream>>>(UU, RA, LC);
  k_rank<<<NRR, 256, 0, stream>>>(LC, SI);
  k_gather<<<NRR, 64, 0, stream>>>(SI, UU, VM, VLS, RA, YS, VMS, VLSS, RBS);

  // G half (sorted coords, upper-triangular only, rows scattered via sort_idx)
  k_ga_acc<1><<<256, 128, 0, stream>>>(YS, YS, RBS, RBS, U_dag, VMS, VLSS,
                                       SI, 0, g_ls, SM, SL, RS);
  // A half (direct coords, full, rows NR..)
  k_ga_acc<0><<<256, 128, 0, stream>>>(UU + (size_t)NRR * 64, UU, RA + NRR, RA,
                                       U_A, VM, VLS, nullptr, NRR, g_ls,
                                       SM, SL, RS);
  // pqz reduction
  k_pqz<<<NALL, 32, 0, stream>>>(ZZ, QZ, SM, SL, RS, ACC);

  // h = relu([z|u] @ W_o1 + b_o1)   [8192x128x512]  (reuses H buffer)
  k_gemm_wmma<2, 1><<<dim3(8, 512), 128, 0, stream>>>(ZZ, UU, W_o1, b_o1, H,
                                                      NALL, DHH, DXX);
  // output head + likelihoods, final objective
  k_outhead<<<NALL, 32, 0, stream>>>(H, W_o2, b_o2, yR, yM, ACC);
  k_final<<<1, 1, 0, stream>>>(ACC, (float*)d_out);
}